// CurriculumLoss_41669772706423
// MI455X (gfx1250) — compile-verified
//
#include <hip/hip_runtime.h>
#include <hip/hip_bf16.h>
#include <math.h>

// ---- shapes fixed by the reference ----
// B = 2048 rows, V = 50257 vocab, N = 1e6 difficulty table, WARMUP = 1000.

typedef float v2f __attribute__((ext_vector_type(2)));
typedef float v8f __attribute__((ext_vector_type(8)));

#define LSE_THREADS 256

// --------------------------------------------------------------------------
// Phase 1: per-row online logsumexp, one 8-wave workgroup per row.
// Streams 412 MB once: bandwidth-bound (~17.7 us at 23.3 TB/s).
// --------------------------------------------------------------------------
__global__ __launch_bounds__(LSE_THREADS)
void row_lse_kernel(const float* __restrict__ logits,
                    const int*   __restrict__ targets,
                    float*       __restrict__ base_loss,
                    int V)
{
    const int row = blockIdx.x;
    const int tid = threadIdx.x;
    const float* __restrict__ xr = logits + (size_t)row * (size_t)V;

    float m = -INFINITY;
    float s = 0.0f;

    // Row start is only 4B aligned in general (V % 4 == 1): scalar head up to
    // the first 16B-aligned element, then b128 vector body, then scalar tail.
    int p = (int)((4u - (uint32_t)(((size_t)row * (size_t)V) & 3u)) & 3u);
    if (p > V) p = V;

    for (int i = tid; i < p; i += LSE_THREADS) {
        float v = xr[i];
        if (v > m) { s *= __expf(m - v); m = v; }
        s += __expf(v - m);
    }

    const float4* __restrict__ x4 = (const float4*)(xr + p);
    const int nv = (V - p) >> 2;
    for (int i = tid; i < nv; i += LSE_THREADS) {
        // Prefetch ~2 strides ahead of the streaming window (global_prefetch_b8).
        if (i + 2 * LSE_THREADS < nv)
            __builtin_prefetch((const void*)(x4 + i + 2 * LSE_THREADS), 0, 0);
        float4 v = x4[i];
        float mx = fmaxf(fmaxf(v.x, v.y), fmaxf(v.z, v.w));
        if (mx > m) { s *= __expf(m - mx); m = mx; }
        s += __expf(v.x - m) + __expf(v.y - m) + __expf(v.z - m) + __expf(v.w - m);
    }

    for (int i = p + (nv << 2) + tid; i < V; i += LSE_THREADS) {
        float v = xr[i];
        if (v > m) { s *= __expf(m - v); m = v; }
        s += __expf(v - m);
    }

    // Block tree reduction of (max, sumexp) pairs.
    __shared__ float sm[LSE_THREADS];
    __shared__ float ss[LSE_THREADS];
    sm[tid] = m;
    ss[tid] = s;
    __syncthreads();
    for (int off = LSE_THREADS / 2; off > 0; off >>= 1) {
        if (tid < off) {
            float m1 = sm[tid],        s1 = ss[tid];
            float m2 = sm[tid + off],  s2 = ss[tid + off];
            float mm = fmaxf(m1, m2);
            sm[tid] = mm;
            ss[tid] = s1 * __expf(m1 - mm) + s2 * __expf(m2 - mm);
        }
        __syncthreads();
    }

    if (tid == 0) {
        float lse = sm[0] + __logf(ss[0]);
        float xt  = xr[targets[row]];
        base_loss[row] = lse - xt;
    }
}

// --------------------------------------------------------------------------
// Phase 2: single wave32. Per-lane sequential accumulation of
//   num = sum bl_i * e_i,  den = sum e_i,   e_i = exp(-(0.9*old + 0.1*bl)*(1-p))
// then an exact-f32 cross-lane reduction via V_WMMA_F32_16X16X4_F32:
//   A (16x4) holds lane partials:   a.x = num, a.y = den
//     lanes 0..15  -> A[M=lane][K=0..1], lanes 16..31 -> A[M=lane-16][K=2..3]
//   B (4x16) selector: B[0][0]=B[2][0]=1 (num col), B[1][1]=B[3][1]=1 (den col)
//   D[m][0] = num_m + num_{m+16},  D[m][1] = den_m + den_{m+16}
// D layout: lane n, VGPR r -> D[M=r (+8 for lanes>=16)][N = n & 15].
// --------------------------------------------------------------------------
__global__ __launch_bounds__(32)
void finalize_kernel(const float* __restrict__ base_loss,
                     const int*   __restrict__ sample_ids,
                     const float* __restrict__ difficulty,
                     const int*   __restrict__ step_ptr,
                     float*       __restrict__ out,
                     int Btot)
{
    const int lane = threadIdx.x;
    const float progress = (float)step_ptr[0] * (1.0f / 1000.0f);  // WARMUP = 1000
    const float scale = 1.0f - progress;

    float num = 0.0f, den = 0.0f;
    for (int i = lane; i < Btot; i += 32) {
        float bl  = base_loss[i];
        float old = difficulty[sample_ids[i]];
        float nd  = 0.9f * old + 0.1f * bl;            // EMA (detached in reference)
        float e   = __expf(-nd * scale);
        num = fmaf(bl, e, num);
        den += e;
    }

    v2f a; a.x = num; a.y = den;
    v2f b;
    b.x = ((lane & 15) == 0) ? 1.0f : 0.0f;  // B[K=0][N] (lanes<16) / B[K=2][N]
    b.y = ((lane & 15) == 1) ? 1.0f : 0.0f;  // B[K=1][N] (lanes<16) / B[K=3][N]
    v8f c = {};
    c = __builtin_amdgcn_wmma_f32_16x16x4_f32(
            /*neg_a=*/false, a, /*neg_b=*/false, b,
            /*c_mod=*/(short)0, c, /*reuse_a=*/false, /*reuse_b=*/false);

    // Sum the 8 row-slots this lane holds (rows M=0..7 or 8..15 of its column).
    float s = c[0] + c[1] + c[2] + c[3] + c[4] + c[5] + c[6] + c[7];

    // Column 0 (lanes 0,16) = num halves; column 1 (lanes 1,17) = den halves.
    float n_tot = __shfl(s, 0, 32)  + __shfl(s, 16, 32);
    float d_tot = __shfl(s, 1, 32)  + __shfl(s, 17, 32);

    if (lane == 0) out[0] = n_tot / d_tot;
}

// --------------------------------------------------------------------------
extern "C" void kernel_launch(void* const* d_in, const int* in_sizes, int n_in,
                              void* d_out, int out_size, void* d_ws, size_t ws_size,
                              hipStream_t stream)
{
    const float* logits     = (const float*)d_in[0];  // [B, V] f32
    const int*   targets    = (const int*)  d_in[1];  // [B] i32
    const int*   sample_ids = (const int*)  d_in[2];  // [B] i32
    const float* difficulty = (const float*)d_in[3];  // [N] f32
    const int*   step_ptr   = (const int*)  d_in[4];  // scalar i32

    const int Btot = in_sizes[1];
    const int V    = in_sizes[0] / Btot;

    float* base_loss = (float*)d_ws;      // Btot floats of scratch
    float* out       = (float*)d_out;     // 1 float

    row_lse_kernel<<<Btot, LSE_THREADS, 0, stream>>>(logits, targets, base_loss, V);
    finalize_kernel<<<1, 32, 0, stream>>>(base_loss, sample_ids, difficulty,
                                          step_ptr, out, Btot);
}